// HardAttODEblock_29566554866000
// MI455X (gfx1250) — compile-verified
//
#include <hip/hip_runtime.h>
#include <stdint.h>

// ---------------- problem constants ----------------
#define E_EDGES     1600000
#define N_NODES     50000
#define DIM         64
#define ND          (N_NODES * DIM)        // 3,200,000
#define TILE        256
#define N_TILES     (E_EDGES / TILE)       // 6250 (exact)
#define EDGE_BLOCKS 2048
#define RANK_LO     319999u                // 0.2*(E-1) = 319999.8
#define RANK_HI     320000u

// ---------------- workspace layout (bytes) ----------------
#define OFF_HHI    0u          // 65536 u32 high-16 histogram
#define OFF_HL0    262144u     // 65536 u32 low-16 histogram (rank lo)
#define OFF_HL1    524288u     // 65536 u32 low-16 histogram (rank hi)
#define OFF_SEL    786432u     // control: [0]=b0 [1]=rem0 [2]=b1 [3]=rem1 [4]=thr(f32)
#define OFF_SUMS   786496u     // N_NODES f32 segment sums
#define ZERO_BYTES 986496u     // zero [0, ZERO_BYTES) each call
#define OFF_W      986496u     // E f32 edge weights
#define OFF_Z      7386496u    // ND f32 z buffer
#define OFF_K      20186496u   // ND f32 k buffer
// total ~33.0 MB

// ---------------- CDNA5 async global->LDS staging ----------------
#if __has_builtin(__builtin_amdgcn_global_load_async_to_lds_b32)
#define HAS_ASYNC_LDS 1
typedef __attribute__((address_space(1))) int GAS1_i32;   // "__device__ int *"
typedef __attribute__((address_space(3))) int LAS3_i32;   // "__shared__ int *"
__device__ __forceinline__ void async_g2l_b32(const void* gptr, void* lptr) {
  // generic->AS via integer cast: global flat addr == AS1 addr; LDS generic
  // addr low 32 bits == LDS offset (ISA 10.2 aperture rules).
  __builtin_amdgcn_global_load_async_to_lds_b32(
      (GAS1_i32*)(uint64_t)(uintptr_t)gptr,
      (LAS3_i32*)(uint32_t)(uintptr_t)lptr,
      0, 0);
}
#else
#define HAS_ASYNC_LDS 0
#endif

#if __has_builtin(__builtin_amdgcn_s_wait_asynccnt)
#define WAIT_ASYNC(n) __builtin_amdgcn_s_wait_asynccnt(n)
#else
#define WAIT_ASYNC(n) asm volatile("s_wait_asynccnt %0" ::"i"(n) : "memory")
#endif

__device__ __forceinline__ void atomic_add_f32(float* p, float v) {
  unsafeAtomicAdd(p, v);   // hardware global_atomic_add_f32
}

// ---------------- quantile: radix select on float bits ----------------
__global__ __launch_bounds__(256) void hist_hi_k(const float* __restrict__ att,
                                                 unsigned* __restrict__ hist) {
  int e = blockIdx.x * 256 + threadIdx.x;
  unsigned key = __float_as_uint(att[e]);   // att >= 0 -> monotone in bits
  atomicAdd(&hist[key >> 16], 1u);
}

// find bucket b and residual rank within bucket for global rank `rank`
__device__ void find_rank2(const unsigned* __restrict__ hist, unsigned rank,
                           unsigned* outB, unsigned* outRem) {
  __shared__ unsigned chunk[256];
  __shared__ unsigned resB, resR;
  const int t = threadIdx.x;
  __syncthreads();                      // protect chunk[] reuse across calls
  unsigned s = 0;
  const unsigned* p = hist + t * 256;
  for (int i = 0; i < 256; i++) s += p[i];
  chunk[t] = s;
  __syncthreads();
  if (t == 0) {
    unsigned run = 0;
    int c = 0;
    for (; c < 255; c++) {
      if (run + chunk[c] > rank) break;
      run += chunk[c];
    }
    unsigned rem = rank - run;
    int b = c * 256;
    for (;;) {
      unsigned cnt = hist[b];
      if (cnt > rem) break;
      rem -= cnt;
      b++;
    }
    resB = (unsigned)b;
    resR = rem;
  }
  __syncthreads();
  *outB = resB;
  *outRem = resR;
}

__global__ __launch_bounds__(256) void select_hi_k(const unsigned* __restrict__ hist,
                                                   unsigned* __restrict__ sel) {
  unsigned b, r;
  find_rank2(hist, RANK_LO, &b, &r);
  if (threadIdx.x == 0) { sel[0] = b; sel[1] = r; }
  find_rank2(hist, RANK_HI, &b, &r);
  if (threadIdx.x == 0) { sel[2] = b; sel[3] = r; }
}

__global__ __launch_bounds__(256) void hist_lo_k(const float* __restrict__ att,
                                                 const unsigned* __restrict__ sel,
                                                 unsigned* __restrict__ h0,
                                                 unsigned* __restrict__ h1) {
  int e = blockIdx.x * 256 + threadIdx.x;
  unsigned key = __float_as_uint(att[e]);
  unsigned hi = key >> 16, lo = key & 0xFFFFu;
  if (hi == sel[0]) atomicAdd(&h0[lo], 1u);
  if (hi == sel[2]) atomicAdd(&h1[lo], 1u);
}

__global__ __launch_bounds__(256) void select_lo_k(const unsigned* __restrict__ h0,
                                                   const unsigned* __restrict__ h1,
                                                   unsigned* __restrict__ sel) {
  unsigned rem0 = sel[1], rem1 = sel[3];
  unsigned b0h = sel[0], b1h = sel[2];
  unsigned b, r;
  find_rank2(h0, rem0, &b, &r);
  unsigned key0 = (b0h << 16) | b;
  find_rank2(h1, rem1, &b, &r);
  unsigned key1 = (b1h << 16) | b;
  if (threadIdx.x == 0) {
    float v0 = __uint_as_float(key0);  // sorted[319999]
    float v1 = __uint_as_float(key1);  // sorted[320000]
    ((float*)sel)[4] = v0 + 0.8f * (v1 - v0);  // numpy 'linear' interpolation
  }
}

// ---------------- edge-weight normalization ----------------
__global__ __launch_bounds__(256) void segsum_k(const float* __restrict__ att,
                                                const int* __restrict__ src,
                                                const unsigned* __restrict__ sel,
                                                float* __restrict__ sums,
                                                float* __restrict__ wbuf) {
  int e = blockIdx.x * 256 + threadIdx.x;
  float thr = ((const float*)sel)[4];
  float a = att[e];
  float m = (a > thr) ? a : 0.0f;
  wbuf[e] = m;
  if (m != 0.0f) atomic_add_f32(&sums[src[e]], m);
}

__global__ __launch_bounds__(256) void wnorm_k(const int* __restrict__ src,
                                               const float* __restrict__ sums,
                                               float* __restrict__ wbuf) {
  int e = blockIdx.x * 256 + threadIdx.x;
  float m = wbuf[e];
  wbuf[e] = m / (sums[src[e]] + 1e-16f);
}

// ---------------- elementwise RK4 helpers ----------------
__global__ __launch_bounds__(256) void negcopy_k(const float* __restrict__ a,
                                                 float* __restrict__ k) {
  int i = blockIdx.x * 256 + threadIdx.x;
  k[i] = -a[i];
}

// acc = (first? x : acc) + ca*k ; if(!last){ z = x + cz*k ; k = -z }
__global__ __launch_bounds__(256) void combine_k(const float* __restrict__ x,
                                                 float* __restrict__ kz,
                                                 float* __restrict__ acc,
                                                 float* __restrict__ zb,
                                                 float ca, float cz,
                                                 int first, int last) {
  int i = blockIdx.x * 256 + threadIdx.x;
  float k = kz[i];
  float xv = x[i];
  float a = first ? fmaf(ca, k, xv) : fmaf(ca, k, acc[i]);
  acc[i] = a;
  if (!last) {
    float z = fmaf(cz, k, xv);
    zb[i] = z;
    kz[i] = -z;   // init next k accumulator = -z (so k += A z gives A z - z)
  }
}

// ---------------- main SpMM: k += A z (k pre-initialized to -z) ----------------
__global__ __launch_bounds__(256) void edge_spmm_k(const int* __restrict__ src,
                                                   const int* __restrict__ dst,
                                                   const float* __restrict__ w,
                                                   const float* __restrict__ z,
                                                   float* __restrict__ k) {
  __shared__ int   sS[2][TILE];
  __shared__ int   sD[2][TILE];
  __shared__ float sV[2][TILE];
  const int tid = threadIdx.x;
  const int G = gridDim.x;
  int t = blockIdx.x;
  if (t >= N_TILES) return;

  // stage tile t into buffer 0
  {
    int e = t * TILE + tid;
#if HAS_ASYNC_LDS
    async_g2l_b32(&src[e], &sS[0][tid]);
    async_g2l_b32(&dst[e], &sD[0][tid]);
    async_g2l_b32(&w[e],   &sV[0][tid]);
#else
    sS[0][tid] = src[e];
    sD[0][tid] = dst[e];
    sV[0][tid] = w[e];
#endif
  }

  int buf = 0;
  for (; t < N_TILES; t += G) {
    const int nxt = t + G;
    const bool has_next = nxt < N_TILES;   // uniform across block
    if (has_next) {
      int e = nxt * TILE + tid;
#if HAS_ASYNC_LDS
      async_g2l_b32(&src[e], &sS[buf ^ 1][tid]);
      async_g2l_b32(&dst[e], &sD[buf ^ 1][tid]);
      async_g2l_b32(&w[e],   &sV[buf ^ 1][tid]);
#else
      sS[buf ^ 1][tid] = src[e];
      sD[buf ^ 1][tid] = dst[e];
      sV[buf ^ 1][tid] = w[e];
#endif
      WAIT_ASYNC(3);   // current buffer's 3 async ops complete; next 3 in flight
    } else {
      WAIT_ASYNC(0);
    }
    __syncthreads();

    // 64 threads per edge (one wave = one edge x 32 features): uniform
    // per-wave edge metadata -> broadcast LDS reads, coalesced 128B vmem.
    const int g = tid >> 6;   // edge group 0..3
    const int f = tid & 63;   // feature 0..63
#pragma unroll 4
    for (int it = 0; it < TILE / 4; it++) {
      const int el = 4 * it + g;
      const float wv = sV[buf][el];
      if (wv != 0.0f) {       // 20% of edges masked: skip gather + atomic
        const int s = sS[buf][el];
        const int d = sD[buf][el];
        atomic_add_f32(&k[d * DIM + f], wv * z[s * DIM + f]);
      }
    }
    __syncthreads();          // all waves done reading before next overwrite
    buf ^= 1;
  }
}

// ---------------- launcher ----------------
extern "C" void kernel_launch(void* const* d_in, const int* in_sizes, int n_in,
                              void* d_out, int out_size, void* d_ws, size_t ws_size,
                              hipStream_t stream) {
  const float* x   = (const float*)d_in[0];
  const float* att = (const float*)d_in[1];
  const int*   ei  = (const int*)d_in[2];
  const int* src = ei;            // edge_index[0]
  const int* dst = ei + E_EDGES;  // edge_index[1]
  float* out = (float*)d_out;
  char* ws = (char*)d_ws;

  unsigned* hhi  = (unsigned*)(ws + OFF_HHI);
  unsigned* hl0  = (unsigned*)(ws + OFF_HL0);
  unsigned* hl1  = (unsigned*)(ws + OFF_HL1);
  unsigned* sel  = (unsigned*)(ws + OFF_SEL);
  float*    sums = (float*)(ws + OFF_SUMS);
  float*    wbuf = (float*)(ws + OFF_W);
  float*    zbuf = (float*)(ws + OFF_Z);
  float*    kbuf = (float*)(ws + OFF_K);

  (void)hipMemsetAsync(ws, 0, ZERO_BYTES, stream);

  // exact 0.2-quantile of att via 2-level radix select on float bits
  hist_hi_k  <<<N_TILES, 256, 0, stream>>>(att, hhi);
  select_hi_k<<<1,       256, 0, stream>>>(hhi, sel);
  hist_lo_k  <<<N_TILES, 256, 0, stream>>>(att, sel, hl0, hl1);
  select_lo_k<<<1,       256, 0, stream>>>(hl0, hl1, sel);

  // masked attention, per-src renormalization
  segsum_k<<<N_TILES, 256, 0, stream>>>(att, src, sel, sums, wbuf);
  wnorm_k <<<N_TILES, 256, 0, stream>>>(src, sums, wbuf);

  const int NB = ND / 256;  // 12500
  // k1 = A x - x
  negcopy_k  <<<NB, 256, 0, stream>>>(x, kbuf);
  edge_spmm_k<<<EDGE_BLOCKS, 256, 0, stream>>>(src, dst, wbuf, x, kbuf);
  // acc = x + k1/6 ; z = x + 0.5 k1 ; k := -z
  combine_k  <<<NB, 256, 0, stream>>>(x, kbuf, out, zbuf, 1.0f/6.0f, 0.5f, 1, 0);
  // k2
  edge_spmm_k<<<EDGE_BLOCKS, 256, 0, stream>>>(src, dst, wbuf, zbuf, kbuf);
  combine_k  <<<NB, 256, 0, stream>>>(x, kbuf, out, zbuf, 1.0f/3.0f, 0.5f, 0, 0);
  // k3
  edge_spmm_k<<<EDGE_BLOCKS, 256, 0, stream>>>(src, dst, wbuf, zbuf, kbuf);
  combine_k  <<<NB, 256, 0, stream>>>(x, kbuf, out, zbuf, 1.0f/3.0f, 1.0f, 0, 0);
  // k4
  edge_spmm_k<<<EDGE_BLOCKS, 256, 0, stream>>>(src, dst, wbuf, zbuf, kbuf);
  combine_k  <<<NB, 256, 0, stream>>>(x, kbuf, out, zbuf, 1.0f/6.0f, 0.0f, 0, 1);
}